// mvdr_atten_20418274525320
// MI455X (gfx1250) — compile-verified
//
#include <hip/hip_runtime.h>

#ifndef __has_builtin
#define __has_builtin(x) 0
#endif

#if __has_builtin(__builtin_amdgcn_global_load_async_to_lds_b128)
#define MVDR_USE_ASYNC 1
#else
#define MVDR_USE_ASYNC 0
#endif

namespace {
constexpr int kB = 2, kT = 600, kF = 513, kM = 5;
constexpr unsigned kNP = (unsigned)kB * kT * kF;   // 615600 points, (b,t,f) flat order
constexpr int kBlock = 128;                        // 4 waves/block
constexpr int kWaves = kBlock / 32;
constexpr unsigned kCorrB = kM * kM * 2 * 4;       // 200 B per point per corr matrix
constexpr unsigned kSpecB = kM * 2 * 4;            // 40 B per point
constexpr unsigned kWCorr = 32 * kCorrB;           // 6400 B per wave per matrix
constexpr unsigned kWSpec = 32 * kSpecB;           // 1280 B per wave
constexpr unsigned kWBytes = 2 * kWCorr + kWSpec;  // 14080 B per wave
}

__device__ __forceinline__ float2 cmul(float2 a, float2 b) {
  return make_float2(a.x * b.x - a.y * b.y, a.x * b.y + a.y * b.x);
}
// acc - a*b (complex), FMA-friendly
__device__ __forceinline__ float2 cnmsub(float2 acc, float2 a, float2 b) {
  return make_float2(acc.x - (a.x * b.x - a.y * b.y),
                     acc.y - (a.x * b.y + a.y * b.x));
}

#if MVDR_USE_ASYNC
// Builtin signature (per hipcc diagnostic): (v4i AS1*, v4i AS3*, imm offset, imm cpol)
typedef int v4i __attribute__((vector_size(16)));
typedef v4i __attribute__((address_space(1))) gv4i;  // global
typedef v4i __attribute__((address_space(3))) lv4i;  // LDS

__device__ __forceinline__ void async_cp16(const void* g, void* l) {
  __builtin_amdgcn_global_load_async_to_lds_b128(
      (gv4i*)g, (lv4i*)l, /*offset=*/0, /*cpol=*/0);
}
__device__ __forceinline__ void wait_async_zero() {
#if __has_builtin(__builtin_amdgcn_s_wait_asynccnt)
  __builtin_amdgcn_s_wait_asynccnt(0);
#else
  asm volatile("s_wait_asynccnt 0" ::: "memory");
#endif
}
#endif

__global__ __launch_bounds__(kBlock) void mvdr_atten_kernel(
    const float2* __restrict__ spec,    // (B,T,F,M) complex
    const float2* __restrict__ corrS,   // (B,T,F,M,M) complex
    const float2* __restrict__ corrN,   // (B,T,F,M,M) complex
    float2* __restrict__ out)           // (B,F,T) complex
{
#if MVDR_USE_ASYNC
  __shared__ char lds[kWaves * kWBytes];
  const unsigned wave = threadIdx.x >> 5;
  const unsigned lane = threadIdx.x & 31;
  char* wlds = &lds[wave * kWBytes];
  {
    // Each wave async-copies its 32 points' packed data (contiguous in global)
    // into its private LDS slab: coalesced b128 per lane, clamped at the tail.
    const unsigned wbase = blockIdx.x * (unsigned)kBlock + wave * 32u;
    const char* gN = (const char*)corrN;
    const char* gS = (const char*)corrS;
    const char* gX = (const char*)spec;
    const unsigned cb = wbase * kCorrB;
    const unsigned xb = wbase * kSpecB;
    const unsigned climit = kNP * kCorrB - 16u;   // 16B-aligned last chunk
    const unsigned xlimit = kNP * kSpecB - 16u;
    for (unsigned c = lane; c < kWCorr / 16u; c += 32u) {  // 400 chunks -> 13 iters
      unsigned g = cb + c * 16u;  if (g > climit) g = climit;
      async_cp16(gN + g, wlds + c * 16u);
      async_cp16(gS + g, wlds + kWCorr + c * 16u);
    }
    for (unsigned c = lane; c < kWSpec / 16u; c += 32u) {  // 80 chunks -> 3 iters
      unsigned g = xb + c * 16u;  if (g > xlimit) g = xlimit;
      async_cp16(gX + g, wlds + 2u * kWCorr + c * 16u);
    }
    wait_async_zero();
  }
#endif

  const unsigned p = blockIdx.x * (unsigned)kBlock + threadIdx.x;
  if (p >= kNP) return;

#if MVDR_USE_ASYNC
  const float2* An = (const float2*)(wlds + lane * kCorrB);
  const float2* As = (const float2*)(wlds + kWCorr + lane * kCorrB);
  const float2* Ax = (const float2*)(wlds + 2u * kWCorr + lane * kSpecB);
#else
  const float2* An = corrN + (size_t)p * (kM * kM);
  const float2* As = corrS + (size_t)p * (kM * kM);
  const float2* Ax = spec  + (size_t)p * kM;
#endif

  // ---- Load Phi_N, add 1e-7*(1+1i)*I ----
  float2 A[kM * kM];
#pragma unroll
  for (int i = 0; i < kM * kM; ++i) A[i] = An[i];
#pragma unroll
  for (int k = 0; k < kM; ++k) { A[k * 6].x += 1e-7f; A[k * 6].y += 1e-7f; }

  // ---- In-place LU (no pivot); Dinv[k] = 1/U[k][k] ----
  float2 Dinv[kM];
#pragma unroll
  for (int k = 0; k < kM; ++k) {
    float2 d = A[k * kM + k];
    float r = 1.0f / (d.x * d.x + d.y * d.y);
    float2 di = make_float2(d.x * r, -d.y * r);
    Dinv[k] = di;
#pragma unroll
    for (int i = k + 1; i < kM; ++i) {
      float2 l = cmul(A[i * kM + k], di);
      A[i * kM + k] = l;
#pragma unroll
      for (int j = k + 1; j < kM; ++j)
        A[i * kM + j] = cnmsub(A[i * kM + j], l, A[k * kM + j]);
    }
  }

  // ---- Solve Phi_N * X = Phi_S column by column; keep diag (trace) & col 0 ----
  float2 tr = make_float2(0.f, 0.f);
  float2 col0[kM];
#pragma unroll
  for (int j = 0; j < kM; ++j) {
    float2 y[kM];
#pragma unroll
    for (int i = 0; i < kM; ++i) {           // forward: L y = b  (unit diag)
      float2 s = As[i * kM + j];
#pragma unroll
      for (int k2 = 0; k2 < i; ++k2) s = cnmsub(s, A[i * kM + k2], y[k2]);
      y[i] = s;
    }
#pragma unroll
    for (int i = kM - 1; i >= 0; --i) {      // backward: U x = y
      float2 s = y[i];
#pragma unroll
      for (int k2 = i + 1; k2 < kM; ++k2) s = cnmsub(s, A[i * kM + k2], y[k2]);
      y[i] = cmul(s, Dinv[i]);
    }
    tr.x += y[j].x; tr.y += y[j].y;          // trace element X[j][j]
    if (j == 0) {
#pragma unroll
      for (int i = 0; i < kM; ++i) col0[i] = y[i];
    }
  }

  // ---- W = col0 / (tr + eps + i*eps);  S_hat = sum conj(W_m) * x_m ----
  const float eps = 1.1920929e-7f;  // FLT_EPSILON
  float2 den = make_float2(tr.x + eps, tr.y + eps);
  float r = 1.0f / (den.x * den.x + den.y * den.y);
  float2 dinv = make_float2(den.x * r, -den.y * r);
  float2 acc = make_float2(0.f, 0.f);
#pragma unroll
  for (int m = 0; m < kM; ++m) {
    float2 W = cmul(col0[m], dinv);
    float2 xm = Ax[m];
    acc.x += W.x * xm.x + W.y * xm.y;   // conj(W)*x
    acc.y += W.x * xm.y - W.y * xm.x;
  }

  // p is (b,t,f) flat; output is (b,f,t)
  const unsigned bb = p / (unsigned)(kT * kF);
  const unsigned rem = p - bb * (unsigned)(kT * kF);
  const unsigned t = rem / (unsigned)kF;
  const unsigned f = rem - t * (unsigned)kF;
  out[(size_t)(bb * kF + f) * kT + t] = acc;
}

extern "C" void kernel_launch(void* const* d_in, const int* in_sizes, int n_in,
                              void* d_out, int out_size, void* d_ws, size_t ws_size,
                              hipStream_t stream) {
  (void)in_sizes; (void)n_in; (void)out_size; (void)d_ws; (void)ws_size;
  const float2* spec  = (const float2*)d_in[0];   // spec_mix_t
  const float2* corrS = (const float2*)d_in[1];   // corr_speech
  const float2* corrN = (const float2*)d_in[2];   // corr_noise
  float2* out = (float2*)d_out;

  const unsigned grid = (kNP + (unsigned)kBlock - 1u) / (unsigned)kBlock;  // 4810
  mvdr_atten_kernel<<<dim3(grid), dim3(kBlock), 0, stream>>>(spec, corrS, corrN, out);
}